// GNNClassifier_GCN_embed_33397665693793
// MI455X (gfx1250) — compile-verified
//
#include <hip/hip_runtime.h>

typedef __attribute__((ext_vector_type(2))) float v2f;
typedef __attribute__((ext_vector_type(8))) float v8f;

#define N_NODES 50000
#define N_EDGES 500000
#define N_FEAT 128
#define EMB_DIM 1024
#define KTOT 1152   // N_FEAT + EMB_DIM
#define HID 128
#define N_GRAPHS 64
#define OUT_DIM 10

// ---------------- small utility kernels ----------------

__global__ void k_fill(float* __restrict__ p, float v, int n) {
  int i = blockIdx.x * blockDim.x + threadIdx.x;
  if (i < n) p[i] = v;
}

__global__ void k_deg_edges(const int* __restrict__ dst, const float* __restrict__ ew,
                            float* __restrict__ deg, int nE) {
  int e = blockIdx.x * blockDim.x + threadIdx.x;
  if (e < nE) atomicAdd(&deg[dst[e]], ew[e]);
}

__global__ void k_dinv(float* __restrict__ deg, int n) {
  int i = blockIdx.x * blockDim.x + threadIdx.x;
  if (i < n) {
    float d = deg[i];
    deg[i] = d > 0.f ? rsqrtf(d) : 0.f;
  }
}

// agg[i,f] = dinv[i]^2 * xw[i,f]   (self-loop term; also initializes agg)
__global__ void k_selfloop(const float* __restrict__ dinv, const float* __restrict__ xw,
                           float* __restrict__ agg, int n) {
  int i = blockIdx.x * blockDim.x + threadIdx.x;
  if (i < n) {
    float dv = dinv[i >> 7];
    agg[i] = dv * dv * xw[i];
  }
}

// one wave32 per edge: lane handles 4 features (32*4 = 128)
__global__ void k_edge_agg(const int* __restrict__ src, const int* __restrict__ dst,
                           const float* __restrict__ ew, const float* __restrict__ dinv,
                           const float* __restrict__ xw, float* __restrict__ agg, int nE) {
  int e = blockIdx.x * (blockDim.x >> 5) + (threadIdx.x >> 5);
  if (e >= nE) return;
  int lane = threadIdx.x & 31;
  int s = src[e], d = dst[e];
  float norm = dinv[s] * ew[e] * dinv[d];
  const float4* xp = (const float4*)(xw + (size_t)s * HID);
  float4 v = xp[lane];
  float* op = agg + (size_t)d * HID + lane * 4;
  atomicAdd(op + 0, norm * v.x);
  atomicAdd(op + 1, norm * v.y);
  atomicAdd(op + 2, norm * v.z);
  atomicAdd(op + 3, norm * v.w);
}

__global__ void k_bias_relu(float* __restrict__ h, const float* __restrict__ b, int n) {
  int i = blockIdx.x * blockDim.x + threadIdx.x;
  if (i < n) {
    float v = h[i] + b[i & (HID - 1)];
    h[i] = v > 0.f ? v : 0.f;
  }
}

// ---------------- WMMA f32 GEMM kernels ----------------
// Layout per ISA 7.12.2:
//  A 16x4 f32: lane l<16 -> M=l, {v0,v1}={K0,K1}; lane l>=16 -> M=l-16, {K2,K3}
//  B 4x16    : lane l<16 -> N=l, {v0,v1}={K0,K1}; lane l>=16 -> N=l-16, {K2,K3}
//  C/D 16x16 : vgpr r, lane l<16 -> (M=r,   N=l); lane l>=16 -> (M=r+8, N=l-16)

// GEMM1 with fused concat: A = [x | emb_table[node_ids]]  (50000 x 1152) @ W1 (1152 x 128)
__global__ __launch_bounds__(256) void k_gemm1_concat(
    const float* __restrict__ x, const float* __restrict__ emb,
    const int* __restrict__ node_ids, const float* __restrict__ W1,
    float* __restrict__ out) {
  __shared__ __align__(16) float As[16][KTOT + 4];  // +4 pad: conflict-free lane banks
  __shared__ int nid_s[16];
  const int tid = threadIdx.x;
  const int rowbase = blockIdx.x << 4;
  if (tid < 16) nid_s[tid] = node_ids[rowbase + tid];
  __syncthreads();
  // stage 16 x 1152 floats = 4608 float4, 18 per thread
  for (int j = 0; j < 18; ++j) {
    int idx = tid + (j << 8);
    int row = idx / 288;            // 288 float4 per row
    int c4 = idx - row * 288;
    const float* sp = (c4 < 32)
        ? (x + (size_t)(rowbase + row) * N_FEAT + (c4 << 2))
        : (emb + (size_t)nid_s[row] * EMB_DIM + ((c4 - 32) << 2));
    *(float4*)&As[row][c4 << 2] = *(const float4*)sp;
  }
  __syncthreads();
  const int wave = tid >> 5, lane = tid & 31;
  const int m = lane & 15;
  const int kh = (lane >> 4) << 1;  // 0 or 2
  const int n0 = wave << 4;
  v8f acc;
#pragma unroll
  for (int r = 0; r < 8; ++r) acc[r] = 0.f;
#pragma unroll 8
  for (int k0 = 0; k0 < KTOT; k0 += 4) {
    v2f a = *(const v2f*)&As[m][k0 + kh];          // ds_load_b64
    const float* bp = W1 + (size_t)(k0 + kh) * HID + n0 + m;
    v2f b;
    b.x = bp[0];
    b.y = bp[HID];
    acc = __builtin_amdgcn_wmma_f32_16x16x4_f32(false, a, false, b, (short)0, acc,
                                                false, false);
  }
  float* op = out + ((size_t)rowbase + (size_t)((lane >> 4) << 3)) * HID + n0 + m;
#pragma unroll
  for (int r = 0; r < 8; ++r) op[(size_t)r * HID] = acc[r];
}

// GEMM2: h1 (50000 x 128) @ W2 (128 x 128)
__global__ __launch_bounds__(256) void k_gemm2(
    const float* __restrict__ A, const float* __restrict__ W,
    float* __restrict__ out) {
  __shared__ __align__(16) float As[16][HID + 4];
  const int tid = threadIdx.x;
  const int rowbase = blockIdx.x << 4;
#pragma unroll
  for (int j = 0; j < 2; ++j) {
    int idx = tid + (j << 8);       // 0..511
    int row = idx >> 5;             // 32 float4 per row
    int c4 = idx & 31;
    *(float4*)&As[row][c4 << 2] =
        *(const float4*)(A + (size_t)(rowbase + row) * HID + (c4 << 2));
  }
  __syncthreads();
  const int wave = tid >> 5, lane = tid & 31;
  const int m = lane & 15;
  const int kh = (lane >> 4) << 1;
  const int n0 = wave << 4;
  v8f acc;
#pragma unroll
  for (int r = 0; r < 8; ++r) acc[r] = 0.f;
#pragma unroll 8
  for (int k0 = 0; k0 < HID; k0 += 4) {
    v2f a = *(const v2f*)&As[m][k0 + kh];
    const float* bp = W + (size_t)(k0 + kh) * HID + n0 + m;
    v2f b;
    b.x = bp[0];
    b.y = bp[HID];
    acc = __builtin_amdgcn_wmma_f32_16x16x4_f32(false, a, false, b, (short)0, acc,
                                                false, false);
  }
  float* op = out + ((size_t)rowbase + (size_t)((lane >> 4) << 3)) * HID + n0 + m;
#pragma unroll
  for (int r = 0; r < 8; ++r) op[(size_t)r * HID] = acc[r];
}

// ---------------- pooling + FC ----------------

__global__ void k_pool(const float* __restrict__ h, const int* __restrict__ batch,
                       float* __restrict__ pooled, int n) {
  int i = blockIdx.x * blockDim.x + threadIdx.x;  // over N*HID
  if (i < n) {
    int node = i >> 7, f = i & (HID - 1);
    atomicAdd(&pooled[batch[node] * HID + f], h[i]);
  }
}

__global__ void k_count(const int* __restrict__ batch, float* __restrict__ counts, int n) {
  int i = blockIdx.x * blockDim.x + threadIdx.x;
  if (i < n) atomicAdd(&counts[batch[i]], 1.0f);
}

__global__ void k_final(const float* __restrict__ pooled, const float* __restrict__ counts,
                        const float* __restrict__ fcW, const float* __restrict__ fcb,
                        float* __restrict__ out) {
  int i = blockIdx.x * blockDim.x + threadIdx.x;
  if (i >= N_GRAPHS * OUT_DIM) return;
  int g = i / OUT_DIM, o = i - g * OUT_DIM;
  float cnt = counts[g];
  cnt = cnt > 1.f ? cnt : 1.f;
  float inv = 1.f / cnt;
  float s = fcb[o];
  for (int f = 0; f < HID; ++f) s += pooled[g * HID + f] * inv * fcW[f * OUT_DIM + o];
  out[i] = s;
}

// ---------------- launcher ----------------

extern "C" void kernel_launch(void* const* d_in, const int* in_sizes, int n_in,
                              void* d_out, int out_size, void* d_ws, size_t ws_size,
                              hipStream_t stream) {
  (void)in_sizes; (void)n_in; (void)out_size; (void)ws_size;
  const float* x    = (const float*)d_in[0];
  const float* ew   = (const float*)d_in[1];
  const float* emb  = (const float*)d_in[2];
  const float* W1   = (const float*)d_in[3];
  const float* b1   = (const float*)d_in[4];
  const float* W2   = (const float*)d_in[5];
  const float* b2   = (const float*)d_in[6];
  const float* fcW  = (const float*)d_in[7];
  const float* fcb  = (const float*)d_in[8];
  const int* eidx   = (const int*)d_in[9];
  const int* batch  = (const int*)d_in[10];
  const int* nids   = (const int*)d_in[11];
  const int* esrc = eidx;
  const int* edst = eidx + N_EDGES;

  float* ws = (float*)d_ws;
  float* deg    = ws;                                   // N (becomes dinv in place)
  float* bufA   = ws + 50048;                           // N*HID (16B aligned)
  float* bufB   = bufA + (size_t)N_NODES * HID;         // N*HID
  float* pooled = bufB + (size_t)N_NODES * HID;         // 64*128
  float* counts = pooled + N_GRAPHS * HID;              // 64

  const int NH = N_NODES * HID;
  const int TB = 256;

  // normalization: deg = 1 (self loop) + sum edge weights -> rsqrt
  k_fill<<<(N_NODES + TB - 1) / TB, TB, 0, stream>>>(deg, 1.0f, N_NODES);
  k_deg_edges<<<(N_EDGES + TB - 1) / TB, TB, 0, stream>>>(edst, ew, deg, N_EDGES);
  k_dinv<<<(N_NODES + TB - 1) / TB, TB, 0, stream>>>(deg, N_NODES);

  // layer 1: fused concat GEMM, then normalized aggregation, bias+relu
  k_gemm1_concat<<<N_NODES / 16, TB, 0, stream>>>(x, emb, nids, W1, bufA);
  k_selfloop<<<(NH + TB - 1) / TB, TB, 0, stream>>>(deg, bufA, bufB, NH);
  k_edge_agg<<<(N_EDGES + 7) / 8, TB, 0, stream>>>(esrc, edst, ew, deg, bufA, bufB, N_EDGES);
  k_bias_relu<<<(NH + TB - 1) / TB, TB, 0, stream>>>(bufB, b1, NH);

  // layer 2
  k_gemm2<<<N_NODES / 16, TB, 0, stream>>>(bufB, W2, bufA);
  k_selfloop<<<(NH + TB - 1) / TB, TB, 0, stream>>>(deg, bufA, bufB, NH);
  k_edge_agg<<<(N_EDGES + 7) / 8, TB, 0, stream>>>(esrc, edst, ew, deg, bufA, bufB, N_EDGES);
  k_bias_relu<<<(NH + TB - 1) / TB, TB, 0, stream>>>(bufB, b2, NH);

  // mean pool per graph + FC
  k_fill<<<(N_GRAPHS * HID + N_GRAPHS + TB - 1) / TB, TB, 0, stream>>>(
      pooled, 0.0f, N_GRAPHS * HID + N_GRAPHS);
  k_pool<<<(NH + TB - 1) / TB, TB, 0, stream>>>(bufB, batch, pooled, NH);
  k_count<<<(N_NODES + TB - 1) / TB, TB, 0, stream>>>(batch, counts, N_NODES);
  k_final<<<(N_GRAPHS * OUT_DIM + TB - 1) / TB, TB, 0, stream>>>(
      pooled, counts, fcW, fcb, (float*)d_out);
}